// GNNFraudDetector_25237227831894
// MI455X (gfx1250) — compile-verified
//
#include <hip/hip_runtime.h>
#include <hip/hip_bf16.h>

typedef float v2f __attribute__((ext_vector_type(2)));
typedef float v8f __attribute__((ext_vector_type(8)));

#define LDS_STRIDE 66  // transposed-W row stride (even -> 8B aligned pairs, odd/64-bank skew)

// ---------------------------------------------------------------------------
// Degree / normalization
// ---------------------------------------------------------------------------
__global__ void deg_init_kernel(float* __restrict__ deg, int n) {
  int i = blockIdx.x * blockDim.x + threadIdx.x;
  if (i < n) deg[i] = 1.0f;  // self-loop contribution
}

__global__ void deg_edge_kernel(const int* __restrict__ col, float* __restrict__ deg, int e) {
  int i = blockIdx.x * blockDim.x + threadIdx.x;
  if (i < e) atomicAdd(&deg[col[i]], 1.0f);
}

__global__ void deg_rsqrt_kernel(float* __restrict__ deg, int n) {
  int i = blockIdx.x * blockDim.x + threadIdx.x;
  if (i < n) deg[i] = rsqrtf(deg[i]);  // deg >= 1 always (self-loops)
}

// ---------------------------------------------------------------------------
// Layer-1 GEMM: H[N x 64] = X[N x 64] @ W[64 x 64], fp32 WMMA 16x16x4.
// 2 waves / block, 16 rows / wave, 4 N-tiles, 16 K-steps (fully unrolled).
// ---------------------------------------------------------------------------
__global__ __launch_bounds__(64) void gemm64_wmma_kernel(const float* __restrict__ X,
                                                         const float* __restrict__ W,
                                                         float* __restrict__ H) {
  __shared__ float Wt[64 * LDS_STRIDE];  // Wt[n][k] = W[k][n]
  for (int p = threadIdx.x; p < 64 * 64; p += 64) {
    int k = p >> 6, n = p & 63;
    Wt[n * LDS_STRIDE + k] = W[p];
  }
  __syncthreads();

  const int wave = threadIdx.x >> 5;
  const int lane = threadIdx.x & 31;
  const int half = lane >> 4;   // which 16-lane half
  const int nc   = lane & 15;   // A-row index / B-col index within tile
  const int rowBase = blockIdx.x * 32 + wave * 16;

  const float* xrow = X + (size_t)(rowBase + nc) * 64;
  v8f acc0 = {}, acc1 = {}, acc2 = {}, acc3 = {};

#pragma unroll
  for (int k0 = 0; k0 < 64; k0 += 4) {
    const int kb = k0 + half * 2;  // lane's K-pair per 16x4 f32 A layout
    v2f a  = *(const v2f*)(xrow + kb);
    v2f b0 = *(const v2f*)(&Wt[(0 * 16 + nc) * LDS_STRIDE + kb]);
    v2f b1 = *(const v2f*)(&Wt[(1 * 16 + nc) * LDS_STRIDE + kb]);
    v2f b2 = *(const v2f*)(&Wt[(2 * 16 + nc) * LDS_STRIDE + kb]);
    v2f b3 = *(const v2f*)(&Wt[(3 * 16 + nc) * LDS_STRIDE + kb]);
    acc0 = __builtin_amdgcn_wmma_f32_16x16x4_f32(false, a, false, b0, (short)0, acc0, false, false);
    acc1 = __builtin_amdgcn_wmma_f32_16x16x4_f32(false, a, false, b1, (short)0, acc1, false, false);
    acc2 = __builtin_amdgcn_wmma_f32_16x16x4_f32(false, a, false, b2, (short)0, acc2, false, false);
    acc3 = __builtin_amdgcn_wmma_f32_16x16x4_f32(false, a, false, b3, (short)0, acc3, false, false);
  }

#pragma unroll
  for (int v = 0; v < 8; ++v) {
    float* hrow = H + (size_t)(rowBase + v + half * 8) * 64;  // C/D: M = v + 8*half, N = nc
    hrow[0 * 16 + nc] = acc0[v];
    hrow[1 * 16 + nc] = acc1[v];
    hrow[2 * 16 + nc] = acc2[v];
    hrow[3 * 16 + nc] = acc3[v];
  }
}

// scalar tail (only launched if N % 32 != 0; N=100000 -> unused)
__global__ void gemm64_tail_kernel(const float* __restrict__ X, const float* __restrict__ W,
                                   float* __restrict__ H, int start, int n) {
  int t = blockIdx.x * blockDim.x + threadIdx.x;
  int r = start + (t >> 6), c = t & 63;
  if (r >= n) return;
  float s = 0.f;
  for (int k = 0; k < 64; ++k) s += X[(size_t)r * 64 + k] * W[k * 64 + c];
  H[(size_t)r * 64 + c] = s;
}

// ---------------------------------------------------------------------------
// Aggregation layer 1: h2 = selfloop + scatter-add of normalized messages.
// self-loop pass doubles as h2 initialization (plain stores, no atomics).
// ---------------------------------------------------------------------------
__global__ void self1_kernel(const float* __restrict__ h, const float* __restrict__ dinv,
                             float* __restrict__ h2, int n) {
  int t = blockIdx.x * blockDim.x + threadIdx.x;  // n*16 threads, 4 floats each
  int node = t >> 4, c4 = (t & 15) * 4;
  if (node >= n) return;
  float s = dinv[node] * dinv[node];
  float4 v = *(const float4*)&h[(size_t)node * 64 + c4];
  float4 o = {v.x * s, v.y * s, v.z * s, v.w * s};
  *(float4*)&h2[(size_t)node * 64 + c4] = o;
}

__global__ void edge1_kernel(const float* __restrict__ h, const int* __restrict__ row,
                             const int* __restrict__ col, const float* __restrict__ dinv,
                             float* __restrict__ h2, int e) {
  int t = blockIdx.x * blockDim.x + threadIdx.x;  // e*16 threads, 4 floats each
  int ed = t >> 4, c4 = (t & 15) * 4;
  if (ed >= e) return;
  int r = row[ed], c = col[ed];
  float nm = dinv[r] * dinv[c];
  float4 v = *(const float4*)&h[(size_t)r * 64 + c4];
  float* dst = &h2[(size_t)c * 64 + c4];
  atomicAdd(dst + 0, v.x * nm);
  atomicAdd(dst + 1, v.y * nm);
  atomicAdd(dst + 2, v.z * nm);
  atomicAdd(dst + 3, v.w * nm);
}

// ---------------------------------------------------------------------------
// Layer-2 GEMM: G[N x 2] = relu(H2 + b1) @ W2[64 x 2], WMMA with N padded to 16.
// bias + ReLU fused into the A-fragment load.
// ---------------------------------------------------------------------------
__global__ __launch_bounds__(64) void gemm2_wmma_kernel(const float* __restrict__ H2,
                                                        const float* __restrict__ B1,
                                                        const float* __restrict__ W2,
                                                        float* __restrict__ G) {
  __shared__ float Wt[16 * LDS_STRIDE];  // Wt[n][k]; cols n>=2 zero-padded
  __shared__ float b1s[64];
  for (int p = threadIdx.x; p < 16 * 64; p += 64) {
    int n = p >> 6, k = p & 63;
    Wt[n * LDS_STRIDE + k] = (n < 2) ? W2[k * 2 + n] : 0.0f;
  }
  b1s[threadIdx.x] = B1[threadIdx.x];
  __syncthreads();

  const int wave = threadIdx.x >> 5;
  const int lane = threadIdx.x & 31;
  const int half = lane >> 4;
  const int nc   = lane & 15;
  const int rowBase = blockIdx.x * 32 + wave * 16;

  const float* hrow = H2 + (size_t)(rowBase + nc) * 64;
  v8f acc = {};

#pragma unroll
  for (int k0 = 0; k0 < 64; k0 += 4) {
    const int kb = k0 + half * 2;
    v2f a;
    a.x = fmaxf(hrow[kb + 0] + b1s[kb + 0], 0.0f);
    a.y = fmaxf(hrow[kb + 1] + b1s[kb + 1], 0.0f);
    v2f b = *(const v2f*)(&Wt[nc * LDS_STRIDE + kb]);
    acc = __builtin_amdgcn_wmma_f32_16x16x4_f32(false, a, false, b, (short)0, acc, false, false);
  }

  if (nc < 2) {  // only cols 0,1 are real; divergence after all WMMAs done
#pragma unroll
    for (int v = 0; v < 8; ++v)
      G[(size_t)(rowBase + v + half * 8) * 2 + nc] = acc[v];
  }
}

__global__ void gemm2_tail_kernel(const float* __restrict__ H2, const float* __restrict__ B1,
                                  const float* __restrict__ W2, float* __restrict__ G,
                                  int start, int n) {
  int t = blockIdx.x * blockDim.x + threadIdx.x;
  int r = start + (t >> 1), c = t & 1;
  if (r >= n) return;
  float s = 0.f;
  for (int k = 0; k < 64; ++k)
    s += fmaxf(H2[(size_t)r * 64 + k] + B1[k], 0.0f) * W2[k * 2 + c];
  G[(size_t)r * 2 + c] = s;
}

// ---------------------------------------------------------------------------
// Aggregation layer 2: out = selfloop + b2 (init stores) + edge scatter-add.
// ---------------------------------------------------------------------------
__global__ void self2_kernel(const float* __restrict__ g, const float* __restrict__ dinv,
                             const float* __restrict__ b2, float* __restrict__ out, int n) {
  int i = blockIdx.x * blockDim.x + threadIdx.x;
  if (i >= n) return;
  float s = dinv[i] * dinv[i];
  out[(size_t)i * 2 + 0] = g[(size_t)i * 2 + 0] * s + b2[0];
  out[(size_t)i * 2 + 1] = g[(size_t)i * 2 + 1] * s + b2[1];
}

__global__ void edge2_kernel(const float* __restrict__ g, const int* __restrict__ row,
                             const int* __restrict__ col, const float* __restrict__ dinv,
                             float* __restrict__ out, int e) {
  int i = blockIdx.x * blockDim.x + threadIdx.x;
  if (i >= e) return;
  int r = row[i], c = col[i];
  float nm = dinv[r] * dinv[c];
  float2 v = *(const float2*)&g[(size_t)r * 2];
  atomicAdd(&out[(size_t)c * 2 + 0], v.x * nm);
  atomicAdd(&out[(size_t)c * 2 + 1], v.y * nm);
}

// ---------------------------------------------------------------------------
extern "C" void kernel_launch(void* const* d_in, const int* in_sizes, int n_in,
                              void* d_out, int out_size, void* d_ws, size_t ws_size,
                              hipStream_t stream) {
  const float* x  = (const float*)d_in[0];
  const int*   ei = (const int*)d_in[1];
  const float* W1 = (const float*)d_in[2];
  const float* b1 = (const float*)d_in[3];
  const float* W2 = (const float*)d_in[4];
  const float* b2 = (const float*)d_in[5];
  float* out = (float*)d_out;

  const int N = in_sizes[0] / 64;
  const int E = in_sizes[1] / 2;
  const int* row = ei;       // sources
  const int* col = ei + E;   // destinations

  // workspace: dinv[Npad] | h[N*64] | h2[N*64] | g[N*2]
  const int Npad = (N + 63) & ~63;
  float* dinv = (float*)d_ws;
  float* h    = dinv + Npad;
  float* h2   = h + (size_t)N * 64;
  float* g    = h2 + (size_t)N * 64;

  const int T = 256;
  // degrees -> dinv
  deg_init_kernel<<<(N + T - 1) / T, T, 0, stream>>>(dinv, N);
  deg_edge_kernel<<<(E + T - 1) / T, T, 0, stream>>>(col, dinv, E);
  deg_rsqrt_kernel<<<(N + T - 1) / T, T, 0, stream>>>(dinv, N);

  // layer 1 GEMM (WMMA main + scalar tail)
  const int mainBlocks = N / 32;
  const int tailStart  = mainBlocks * 32;
  if (mainBlocks > 0)
    gemm64_wmma_kernel<<<mainBlocks, 64, 0, stream>>>(x, W1, h);
  if (tailStart < N) {
    int tcnt = (N - tailStart) * 64;
    gemm64_tail_kernel<<<(tcnt + T - 1) / T, T, 0, stream>>>(x, W1, h, tailStart, N);
  }

  // layer 1 aggregation (self-loops initialize h2, then edge atomics)
  self1_kernel<<<((long long)N * 16 + T - 1) / T, T, 0, stream>>>(h, dinv, h2, N);
  edge1_kernel<<<((long long)E * 16 + T - 1) / T, T, 0, stream>>>(h, row, col, dinv, h2, E);

  // layer 2 GEMM (fused bias+ReLU on A; WMMA main + scalar tail)
  if (mainBlocks > 0)
    gemm2_wmma_kernel<<<mainBlocks, 64, 0, stream>>>(h2, b1, W2, g);
  if (tailStart < N) {
    int tcnt = (N - tailStart) * 2;
    gemm2_tail_kernel<<<(tcnt + T - 1) / T, T, 0, stream>>>(h2, b1, W2, g, tailStart, N);
  }

  // layer 2 aggregation (self-loops + b2 initialize out, then edge atomics)
  self2_kernel<<<(N + T - 1) / T, T, 0, stream>>>(g, dinv, b2, out, N);
  edge2_kernel<<<(E + T - 1) / T, T, 0, stream>>>(g, row, col, dinv, out, E);
}